// GPM_80968723464308
// MI455X (gfx1250) — compile-verified
//
#include <hip/hip_runtime.h>
#include <math.h>

#define KK 16
#define INDUCE_RATE 50
#define GP_VAR 1.0f
#define GP_LS 10.0f
#define GP_JITTER 1e-3f
#define CHUNK 256
#define NEG_INV2L2 (-1.0f/(2.0f*GP_LS*GP_LS))

typedef float v2f __attribute__((ext_vector_type(2)));
typedef float v8f __attribute__((ext_vector_type(8)));

__device__ __forceinline__ float wave_max(float v){
#pragma unroll
  for (int o = 16; o > 0; o >>= 1) v = fmaxf(v, __shfl_xor(v, o, 32));
  return v;
}
__device__ __forceinline__ float wave_sum(float v){
#pragma unroll
  for (int o = 16; o > 0; o >>= 1) v += __shfl_xor(v, o, 32);
  return v;
}

// ---------------- K_uu = RBF(T_ind, T_ind) + jitter*I ----------------
__global__ void __launch_bounds__(256) build_kuu(const float* __restrict__ T,
                                                 float* __restrict__ A, int M)
{
  size_t idx = (size_t)blockIdx.x * 256 + threadIdx.x;
  size_t total = (size_t)M * M;
  if (idx >= total) return;
  int i = (int)(idx / M), j = (int)(idx % M);
  float d = T[(size_t)i * INDUCE_RATE] - T[(size_t)j * INDUCE_RATE];
  float v = GP_VAR * expf(d * d * NEG_INV2L2);
  if (i == j) v += GP_JITTER;
  A[idx] = v;
}

// ================= Blocked Gauss-Jordan inverse (B = 16) ================
// Per block step kb (pivot cols c0 = 16*kb):
//   1) bgj_pivot : invert 16x16 diag block in LDS, accumulate logdet(K_uu)
//   2) bgj_panel : R = Dinv * A[block rows, :] (pivot cols <- Dinv);
//                  stage old column panel F = A[:, pivot cols]
//   3) bgj_update: A <- A - F * R for all non-block rows (WMMA GEMM),
//                  pivot-col tiles start from 0 (in-place GJ semantics)

__global__ void __launch_bounds__(32) bgj_pivot(float* __restrict__ A,
                                                float* __restrict__ Dinv,
                                                float* __restrict__ ldK,
                                                int M, int kb)
{
  __shared__ float Dl[256];
  __shared__ float fcol[16];
  __shared__ float psh;
  const int lane = threadIdx.x;
  const int c0 = kb * 16;
  if (lane < 16){
#pragma unroll
    for (int r = 0; r < 16; ++r)
      Dl[r*16 + lane] = A[(size_t)(c0 + r) * M + c0 + lane];
  }
  __syncthreads();
  float ld = 0.f;
  for (int k = 0; k < 16; ++k){
    if (lane == 0) psh = Dl[k*16 + k];
    __syncthreads();
    const float p = psh;
    if (lane == 0) ld += logf(p);
    if (lane < 16){
      float val = (lane == k) ? 1.f : Dl[k*16 + lane];
      Dl[k*16 + lane] = val / p;
    }
    __syncthreads();
    if (lane < 16){
      fcol[lane] = Dl[lane*16 + k];
      if (lane != k) Dl[lane*16 + k] = 0.f;
    }
    __syncthreads();
    if (lane < 16){
#pragma unroll
      for (int i = 0; i < 16; ++i){
        if (i == k) continue;
        Dl[i*16 + lane] = fmaf(-fcol[i], Dl[k*16 + lane], Dl[i*16 + lane]);
      }
    }
    __syncthreads();
  }
  if (lane < 16){
#pragma unroll
    for (int r = 0; r < 16; ++r) Dinv[r*16 + lane] = Dl[r*16 + lane];
  }
  if (lane == 0) ldK[0] = (kb == 0 ? 0.f : ldK[0]) + ld;
}

__global__ void __launch_bounds__(256) bgj_panel(float* __restrict__ A,
                                                 const float* __restrict__ Dinv,
                                                 float* __restrict__ F,
                                                 int M, int kb)
{
  __shared__ float sD[256];
  const int tid = threadIdx.x;
  sD[tid] = Dinv[tid];
  __syncthreads();
  const int j = blockIdx.x * 256 + tid;
  if (j >= M) return;
  const int c0 = kb * 16;
  // stage old column panel (block rows of F are garbage but never used)
#pragma unroll
  for (int c = 0; c < 16; ++c)
    F[(size_t)j * 16 + c] = A[(size_t)j * M + c0 + c];
  // row panel
  if (j >= c0 && j < c0 + 16){
#pragma unroll
    for (int r = 0; r < 16; ++r)
      A[(size_t)(c0 + r) * M + j] = sD[r*16 + (j - c0)];
  } else {
    float tmp[16];
#pragma unroll
    for (int r = 0; r < 16; ++r) tmp[r] = A[(size_t)(c0 + r) * M + j];
#pragma unroll
    for (int r = 0; r < 16; ++r){
      float s = 0.f;
#pragma unroll
      for (int k = 0; k < 16; ++k) s = fmaf(sD[r*16 + k], tmp[k], s);
      A[(size_t)(c0 + r) * M + j] = s;
    }
  }
}

__global__ void __launch_bounds__(32) bgj_update(float* __restrict__ A,
                                                 const float* __restrict__ F,
                                                 int M, int kb)
{
  const int rt = blockIdx.y;
  if (rt == kb) return;                 // block rows already hold R
  const int lane = threadIdx.x;
  const int l = lane & 15;
  const bool hi = lane >= 16;
  const int i0 = rt * 16;
  const int j0 = blockIdx.x * 16;
  const int c0 = kb * 16;
  // C tile (pivot-col tile starts from zero per in-place GJ)
  v8f acc;
#pragma unroll
  for (int r = 0; r < 8; ++r){
    int row = i0 + r + (hi ? 8 : 0);
    acc[r] = (j0 == c0) ? 0.f : A[(size_t)row * M + j0 + l];
  }
  // acc -= F_tile(16x16) x R_tile(16x16) via 4 k-slices of WMMA f32 16x16x4
#pragma unroll
  for (int k0 = 0; k0 < 16; k0 += 4){
    const int ka = k0 + (hi ? 2 : 0);
    v2f Aop;
    Aop.x = -F[(size_t)(i0 + l) * 16 + ka];
    Aop.y = -F[(size_t)(i0 + l) * 16 + ka + 1];
    v2f Bop;
    Bop.x = A[(size_t)(c0 + ka)     * M + j0 + l];
    Bop.y = A[(size_t)(c0 + ka + 1) * M + j0 + l];
    acc = __builtin_amdgcn_wmma_f32_16x16x4_f32(
        false, Aop, false, Bop, (short)0, acc, false, false);
  }
#pragma unroll
  for (int r = 0; r < 8; ++r){
    int row = i0 + r + (hi ? 8 : 0);
    A[(size_t)row * M + j0 + l] = acc[r];
  }
}

// -------- Fallback: single-workgroup in-place GJ (M not multiple of 16)
__global__ void __launch_bounds__(1024) gj_inverse(float* __restrict__ A,
                                                   float* __restrict__ ldk_out,
                                                   int M)
{
  __shared__ float rowbuf[2048];
  __shared__ float colbuf[2048];
  __shared__ float pshare;
  const int tid = threadIdx.x;
  float ld = 0.f;
  for (int k = 0; k < M; ++k){
    if (tid == 0) pshare = A[(size_t)k * M + k];
    __syncthreads();
    const float p  = pshare;
    const float ip = 1.f / p;
    if (tid == 0) ld += logf(p);
    for (int j = tid; j < M; j += 1024){
      float val = (j == k) ? 1.f : A[(size_t)k * M + j];
      val *= ip;
      A[(size_t)k * M + j] = val;
      rowbuf[j] = val;
    }
    for (int i = tid; i < M; i += 1024){
      if (i != k){ colbuf[i] = A[(size_t)i * M + k]; A[(size_t)i * M + k] = 0.f; }
    }
    __syncthreads();
    for (int i = 0; i < M; ++i){
      if (i == k) continue;
      const float f = colbuf[i];
      for (int j = tid; j < M; j += 1024)
        A[(size_t)i * M + j] = fmaf(-f, rowbuf[j], A[(size_t)i * M + j]);
    }
    __syncthreads();
  }
  if (tid == 0) ldk_out[0] = ld;   // logdet(K_uu)
}

// ---------------- y = Kinv @ qmu ; w = Kinv @ u (u = qmu + e^{qs/2} eps)
__global__ void __launch_bounds__(256) inv_matvec(const float* __restrict__ A,
                                                  const float* __restrict__ qmu,
                                                  const float* __restrict__ qs,
                                                  const float* __restrict__ eps,
                                                  float* __restrict__ y,
                                                  float* __restrict__ w, int M)
{
  __shared__ float r1[256], r2[256];
  const int row = blockIdx.x, tid = threadIdx.x;
  float sy = 0.f, sw = 0.f;
  for (int j = tid; j < M; j += 256){
    float a = A[(size_t)row * M + j];
    float q = qmu[j];
    float u = fmaf(expf(0.5f * qs[j]), eps[j], q);
    sy = fmaf(a, q, sy);
    sw = fmaf(a, u, sw);
  }
  r1[tid] = sy; r2[tid] = sw; __syncthreads();
  for (int o = 128; o > 0; o >>= 1){
    if (tid < o){ r1[tid] += r1[tid + o]; r2[tid] += r2[tid + o]; }
    __syncthreads();
  }
  if (tid == 0){ y[row] = r1[0]; w[row] = r2[0]; }
}

// ---------------- KL scalar (ldK = logdet(K_uu)) ----------------
__global__ void __launch_bounds__(256) kl_reduce(const float* __restrict__ A,
                                                 const float* __restrict__ qmu,
                                                 const float* __restrict__ qs,
                                                 const float* __restrict__ y,
                                                 const float* __restrict__ ldK,
                                                 float* __restrict__ klv, int M)
{
  __shared__ float r1[256], r2[256], r3[256];
  const int tid = threadIdx.x;
  float tr = 0.f, qd = 0.f, sq = 0.f;
  for (int m = tid; m < M; m += 256){
    tr = fmaf(A[(size_t)m * M + m], expf(qs[m]), tr);
    qd = fmaf(qmu[m], y[m], qd);
    sq += qs[m];
  }
  r1[tid] = tr; r2[tid] = qd; r3[tid] = sq; __syncthreads();
  for (int o = 128; o > 0; o >>= 1){
    if (tid < o){ r1[tid] += r1[tid+o]; r2[tid] += r2[tid+o]; r3[tid] += r3[tid+o]; }
    __syncthreads();
  }
  // kl = 0.5*(tr + quad - M - logdet(Kinv) - sum qs), -logdet(Kinv)=+logdet(Kuu)
  if (tid == 0) klv[0] = 0.5f * (r1[0] + r2[0] - (float)M + ldK[0] - r3[0]);
}

// ---------------- X[t] = sum_m rbf(T[t], Tind[m]) * w[m] ----------------
__global__ void __launch_bounds__(256) x_project(const float* __restrict__ T,
                                                 const float* __restrict__ w,
                                                 float* __restrict__ X, int n, int M)
{
  __shared__ float sT[256], sW[256];
  const int t = blockIdx.x * 256 + threadIdx.x;
  const float Tt = T[(t < n) ? t : (n - 1)];
  float acc = 0.f;
  for (int m0 = 0; m0 < M; m0 += 256){
    __syncthreads();
    int m = m0 + threadIdx.x;
    sT[threadIdx.x] = (m < M) ? T[(size_t)m * INDUCE_RATE] : 0.f;
    sW[threadIdx.x] = (m < M) ? w[m] : 0.f;
    __syncthreads();
    int cnt = (M - m0 < 256) ? (M - m0) : 256;
    for (int i = 0; i < cnt; ++i){
      float d = Tt - sT[i];
      acc = fmaf(sW[i], expf(d * d * NEG_INV2L2), acc);
    }
  }
  if (t < n) X[t] = GP_VAR * acc;
}

// ---------------- HMM chunked scan (WMMA matrix-product scan) -----------
__global__ void __launch_bounds__(32) hmm_chunk(const float* __restrict__ X,
                                                const float* __restrict__ dT,
                                                const float* __restrict__ y_ll,
                                                const float* __restrict__ bias_A,
                                                const float* __restrict__ bias_ab,
                                                const float* __restrict__ W_pi,
                                                const float* __restrict__ W_ab,
                                                float* __restrict__ chunkM,
                                                float* __restrict__ chunkS, int n)
{
  __shared__ float prod[256];    // running product, row-major
  __shared__ float sbiasA[256];
  __shared__ float swpi[16];
  __shared__ float sP[16*17];    // softmax probs, padded stride 17
  __shared__ float lvec[16];     // scaled emission likelihoods
  const int lane = threadIdx.x;
  const int l = lane & 15;
  const bool hi = lane >= 16;
  const int g = blockIdx.x;
  int s0 = g * CHUNK;
  int s1 = s0 + CHUNK; if (s1 > n - 1) s1 = n - 1;

  for (int i = lane; i < 256; i += 32){
    prod[i] = ((i >> 4) == (i & 15)) ? 1.f : 0.f;
    sbiasA[i] = bias_A[i];
  }
  float wab0 = 0.f, wab1 = 0.f, bab0 = 0.f, bab1 = 0.f;
  if (lane < 16){
    swpi[lane] = W_pi[lane];
    wab0 = W_ab[lane*2+0]; wab1 = W_ab[lane*2+1];
    bab0 = bias_ab[lane*2+0]; bab1 = bias_ab[lane*2+1];
  }
  float scale = 0.f;
  __syncthreads();

  for (int s = s0; s < s1; ++s){
    const float Xs  = X[s];
    const float Xs1 = X[s+1];
    // Gamma + external log-likelihood at time s+1 (lane = class j)
    float llj = -INFINITY;
    if (lane < 16){
      float la = fmaf(Xs1, wab0, bab0);
      float lb = fmaf(Xs1, wab1, bab1);
      float a = expf(la), b = expf(lb);
      float d = dT[s+1];
      llj = a*lb + (a - 1.f)*logf(d) - b*d - lgammaf(a)
          + y_ll[(size_t)(s+1)*KK + lane];
    }
    float c = wave_max(llj);
    scale += c;
    if (lane < 16) lvec[lane] = expf(llj - c);
    // Transition softmax row (lane = row i); store probs to LDS
    if (lane < 16){
      float m = -INFINITY; float vals[16];
#pragma unroll
      for (int j = 0; j < 16; ++j){
        float v = fmaf(Xs, swpi[j], sbiasA[lane*16 + j]);
        vals[j] = v; m = fmaxf(m, v);
      }
      float ssum = 0.f;
#pragma unroll
      for (int j = 0; j < 16; ++j){ float e = expf(vals[j] - m); vals[j] = e; ssum += e; }
      float rinv = 1.f / ssum;
#pragma unroll
      for (int j = 0; j < 16; ++j) sP[lane*17 + j] = vals[j] * rinv;
    }
    __syncthreads();

    const float lj = lvec[l];
    v8f acc = {};
#pragma unroll
    for (int k0 = 0; k0 < 16; k0 += 4){
      // A operand: 16x4 slice of running product (cols k0..k0+3)
      const int ka = k0 + (hi ? 2 : 0);
      v2f Aop; Aop.x = prod[l*16 + ka]; Aop.y = prod[l*16 + ka + 1];
      // B operand: 4x16 slice of M_s = P * diag(lvec), rows k0..k0+3
      const int kr = k0 + (hi ? 2 : 0);
      v2f Bop;
      Bop.x = sP[kr*17 + l]       * lj;
      Bop.y = sP[(kr+1)*17 + l]   * lj;
      acc = __builtin_amdgcn_wmma_f32_16x16x4_f32(
          false, Aop, false, Bop, (short)0, acc, false, false);
    }
    // Rescale so max entry == 1; accumulate log scale
    float pm = 0.f;
#pragma unroll
    for (int r = 0; r < 8; ++r) pm = fmaxf(pm, acc[r]);
    pm = wave_max(pm);
    pm = fmaxf(pm, 1e-30f);
    const float ipm = 1.f / pm;
    scale += logf(pm);
    __syncthreads();
#pragma unroll
    for (int r = 0; r < 8; ++r){
      int row = r + (hi ? 8 : 0);
      prod[row*16 + l] = acc[r] * ipm;
    }
    __syncthreads();
  }
  for (int i = lane; i < 256; i += 32) chunkM[(size_t)g*256 + i] = prod[i];
  if (lane == 0) chunkS[g] = scale;
}

// ---------------- Sequential fold of chunk matrices -> loss ------------
__global__ void __launch_bounds__(32) hmm_combine(const float* __restrict__ X,
                                                  const float* __restrict__ dT,
                                                  const float* __restrict__ y_ll,
                                                  const float* __restrict__ pi,
                                                  const float* __restrict__ W_ab,
                                                  const float* __restrict__ bias_ab,
                                                  const float* __restrict__ chunkM,
                                                  const float* __restrict__ chunkS,
                                                  const float* __restrict__ klv,
                                                  const float* __restrict__ y_loss,
                                                  float* __restrict__ loss_out, int G)
{
  __shared__ float v[16];
  const int lane = threadIdx.x;
  float p = (lane < 16) ? pi[lane] : -INFINITY;
  float pm = wave_max(p);
  float pe = (lane < 16) ? expf(p - pm) : 0.f;
  float lse = pm + logf(wave_sum(pe));
  float ll0 = -INFINITY;
  if (lane < 16){
    float x0 = X[0], d = dT[0];
    float la = fmaf(x0, W_ab[lane*2+0], bias_ab[lane*2+0]);
    float lb = fmaf(x0, W_ab[lane*2+1], bias_ab[lane*2+1]);
    float a = expf(la), b = expf(lb);
    ll0 = a*lb + (a - 1.f)*logf(d) - b*d - lgammaf(a) + y_ll[lane];
  }
  float a0 = (lane < 16) ? (p - lse + ll0) : -INFINITY;
  float c0 = wave_max(a0);
  float S = c0;
  if (lane < 16) v[lane] = expf(a0 - c0);
  __syncthreads();
  for (int g = 0; g < G; ++g){
    float acc = 0.f;
    if (lane < 16){
#pragma unroll 4
      for (int i = 0; i < 16; ++i)
        acc = fmaf(v[i], chunkM[(size_t)g*256 + i*16 + lane], acc);
    }
    float m = wave_max((lane < 16) ? acc : 0.f);
    m = fmaxf(m, 1e-30f);
    S += logf(m) + chunkS[g];
    __syncthreads();
    if (lane < 16) v[lane] = acc / m;
    __syncthreads();
  }
  float sum = wave_sum((lane < 16) ? v[lane] : 0.f);
  if (lane == 0) loss_out[0] = -(logf(sum) + S) + y_loss[0] + klv[0];
}

// ---------------- Orchestration ----------------
extern "C" void kernel_launch(void* const* d_in, const int* in_sizes, int n_in,
                              void* d_out, int out_size, void* d_ws, size_t ws_size,
                              hipStream_t stream)
{
  (void)n_in; (void)out_size; (void)ws_size;
  const float* dT     = (const float*)d_in[0];
  const float* T      = (const float*)d_in[1];
  const float* y_ll   = (const float*)d_in[2];
  const float* y_loss = (const float*)d_in[3];
  const float* qmu    = (const float*)d_in[4];
  const float* qs     = (const float*)d_in[5];
  const float* eps    = (const float*)d_in[6];
  const float* biasA  = (const float*)d_in[7];
  const float* biasab = (const float*)d_in[8];
  const float* Wpi    = (const float*)d_in[9];
  const float* Wab    = (const float*)d_in[10];
  const float* pi     = (const float*)d_in[11];

  const int n = in_sizes[0];      // 100000
  const int M = in_sizes[4];      // 2000 inducing points
  const int G = (n - 1 + CHUNK - 1) / CHUNK;

  float* X    = (float*)d_out;    // X: n floats
  float* loss = X + n;            // loss scalar at the end

  float* W    = (float*)d_ws;
  float* A    = W;                          // M*M (K_uu -> K_uu^-1 in place)
  float* yv   = A + (size_t)M * M;          // M
  float* wv   = yv + M;                     // M
  float* ldK  = wv + M;                     // 1 : logdet(K_uu)
  float* klv  = ldK + 1;                    // 1
  float* Dinv = klv + 1;                    // 256
  float* Fpan = Dinv + 256;                 // M*16
  float* chS  = Fpan + (size_t)M * 16;      // G
  float* chM  = chS + G;                    // G*256

  {
    size_t total = (size_t)M * M;
    int blocks = (int)((total + 255) / 256);
    build_kuu<<<blocks, 256, 0, stream>>>(T, A, M);
  }
  if ((M & 15) == 0){
    const int NT = M / 16;
    for (int kb = 0; kb < NT; ++kb){
      bgj_pivot<<<1, 32, 0, stream>>>(A, Dinv, ldK, M, kb);
      bgj_panel<<<(M + 255) / 256, 256, 0, stream>>>(A, Dinv, Fpan, M, kb);
      bgj_update<<<dim3(NT, NT), 32, 0, stream>>>(A, Fpan, M, kb);
    }
  } else {
    gj_inverse<<<1, 1024, 0, stream>>>(A, ldK, M);
  }
  inv_matvec<<<M, 256, 0, stream>>>(A, qmu, qs, eps, yv, wv, M);
  kl_reduce<<<1, 256, 0, stream>>>(A, qmu, qs, yv, ldK, klv, M);
  x_project<<<(n + 255) / 256, 256, 0, stream>>>(T, wv, X, n, M);
  hmm_chunk<<<G, 32, 0, stream>>>(X, dT, y_ll, biasA, biasab, Wpi, Wab, chM, chS, n);
  hmm_combine<<<1, 32, 0, stream>>>(X, dT, y_ll, pi, Wab, biasab, chM, chS, klv,
                                    y_loss, loss, G);
}